// GNN_9225589752460
// MI455X (gfx1250) — compile-verified
//
#include <hip/hip_runtime.h>
#include <math.h>

// ---------------------------------------------------------------------------
// 2-layer GCN (DGL GraphConv, norm='both') for MI455X / gfx1250 (wave32).
//   h1 = relu((segment_sum((x*no)[src], dst) * ni) @ W1 + b1)
//   y  =      (segment_sum((h1*no)[src], dst) * ni) @ W2 + b2
// no = 1/sqrt(out-degree via src), ni = 1/sqrt(in-degree via dst).
//
// Strategy:
//  * scatter: one lane per (edge, channel); 32 lanes = 1 edge -> coalesced
//    128B gather + 128B of f32 atomics into L2 (all data L2-resident: 192MB).
//  * GEMM: V_WMMA_F32_16X16X4_F32, one wave per 16x32 output tile, with the
//    norm_in pre-scale, bias, relu and next-layer norm_out post-scale fused
//    into the WMMA prologue/epilogue.
// ---------------------------------------------------------------------------

typedef __attribute__((ext_vector_type(2))) float v2f;
typedef __attribute__((ext_vector_type(8))) float v8f;

#define DF 32  // feature dim for all layers (D_IN == D_HID == D_OUT == 32)

__global__ __launch_bounds__(256) void k_zero(float* __restrict__ p, int n) {
  int i = blockIdx.x * blockDim.x + threadIdx.x;
  if (i < n) p[i] = 0.0f;
}

// Accumulate float degrees (exact: counts < 2^24).
__global__ __launch_bounds__(256) void k_degrees(const int* __restrict__ src,
                                                 const int* __restrict__ dst,
                                                 float* __restrict__ deg_out,
                                                 float* __restrict__ deg_in,
                                                 int e) {
  int i = blockIdx.x * blockDim.x + threadIdx.x;
  if (i < e) {
    atomicAdd(&deg_out[src[i]], 1.0f);
    atomicAdd(&deg_in[dst[i]], 1.0f);
  }
}

// In-place deg -> inverse-sqrt norm (0 for zero-degree nodes).
__global__ __launch_bounds__(256) void k_invsqrt(float* __restrict__ a,
                                                 float* __restrict__ b, int n) {
  int i = blockIdx.x * blockDim.x + threadIdx.x;
  if (i < n) {
    float d = a[i];
    a[i] = (d > 0.0f) ? (1.0f / sqrtf(d)) : 0.0f;
    d = b[i];
    b[i] = (d > 0.0f) ? (1.0f / sqrtf(d)) : 0.0f;
  }
}

// xs[i,k] = x[i,k] * nrm[i]
__global__ __launch_bounds__(256) void k_prescale(const float* __restrict__ x,
                                                  const float* __restrict__ nrm,
                                                  float* __restrict__ xs,
                                                  int total) {
  int i = blockIdx.x * blockDim.x + threadIdx.x;
  if (i < total) xs[i] = x[i] * nrm[i >> 5];
}

// One lane per (edge, channel): agg[dst[e], k] += xs[src[e], k]
__global__ __launch_bounds__(256) void k_scatter(const float* __restrict__ xs,
                                                 const int* __restrict__ src,
                                                 const int* __restrict__ dst,
                                                 float* __restrict__ agg,
                                                 int total) {
  int i = blockIdx.x * blockDim.x + threadIdx.x;
  if (i >= total) return;
  int e = i >> 5;
  int k = i & 31;
  int s = src[e];
  int d = dst[e];
  atomicAdd(&agg[(size_t)d * DF + k], xs[(size_t)s * DF + k]);
}

// out[row0:row0+16, 0:32] = op((agg*ni) @ W + b) * post_scale
// using V_WMMA_F32_16X16X4_F32, one wave per 16-row tile, 8 k-steps x 2 n-tiles.
//
// ISA layouts (wave32, 32-bit elements):
//   A 16x4:  lane L (L<16): V0=K0,V1=K1 of row M=L; lane L>=16: V0=K2,V1=K3.
//   B 4x16:  V0: K = 2*(lane>>4),   N = lane&15;  V1: K = 2*(lane>>4)+1.
//   C/D:     Vv: M = v + 8*(lane>>4), N = lane&15.
__global__ __launch_bounds__(256) void k_gemm_wmma(
    const float* __restrict__ agg, const float* __restrict__ norm_in,
    const float* __restrict__ W, const float* __restrict__ bias,
    const float* __restrict__ post_scale,  // nullptr => no post scale
    float* __restrict__ out, int n, int do_relu) {
  const int lane = threadIdx.x & 31;
  const int wave = threadIdx.x >> 5;
  const int row0 = (blockIdx.x * 8 + wave) * 16;
  if (row0 >= n) return;  // wave-uniform: EXEC stays all-ones for WMMA

  const int mrow = lane & 15;   // row within tile (A) / column (B,C,D)
  const int khalf = lane >> 4;  // 0: K0/K1 half, 1: K2/K3 half
  int rload = row0 + mrow;
  if (rload > n - 1) rload = n - 1;  // clamp loads for partial tail tile
  const float ni = norm_in[rload];
  const float* __restrict__ arow = agg + (size_t)rload * DF;
  const int nb = lane & 15;

  v8f acc0 = {};  // columns 0..15
  v8f acc1 = {};  // columns 16..31
#pragma unroll
  for (int k0 = 0; k0 < DF; k0 += 4) {
    const int ka = k0 + khalf * 2;
    v2f a;
    a.x = arow[ka + 0] * ni;
    a.y = arow[ka + 1] * ni;
    v2f b0, b1;
    b0.x = W[(ka + 0) * DF + nb];
    b0.y = W[(ka + 1) * DF + nb];
    b1.x = W[(ka + 0) * DF + 16 + nb];
    b1.y = W[(ka + 1) * DF + 16 + nb];
    acc0 = __builtin_amdgcn_wmma_f32_16x16x4_f32(false, a, false, b0,
                                                 (short)0, acc0, false, false);
    acc1 = __builtin_amdgcn_wmma_f32_16x16x4_f32(false, a, false, b1,
                                                 (short)0, acc1, false, false);
  }

  const float bc0 = bias[nb];
  const float bc1 = bias[16 + nb];
#pragma unroll
  for (int v = 0; v < 8; ++v) {
    const int m = row0 + v + 8 * khalf;
    if (m < n) {
      float o0 = acc0[v] + bc0;
      float o1 = acc1[v] + bc1;
      if (do_relu) {
        o0 = fmaxf(o0, 0.0f);
        o1 = fmaxf(o1, 0.0f);
      }
      const float s = post_scale ? post_scale[m] : 1.0f;
      out[(size_t)m * DF + nb] = o0 * s;
      out[(size_t)m * DF + 16 + nb] = o1 * s;
    }
  }
}

extern "C" void kernel_launch(void* const* d_in, const int* in_sizes, int n_in,
                              void* d_out, int out_size, void* d_ws,
                              size_t ws_size, hipStream_t stream) {
  const float* x = (const float*)d_in[0];   // [N,32]
  const int* src = (const int*)d_in[1];     // [E]
  const int* dst = (const int*)d_in[2];     // [E]
  const float* W1 = (const float*)d_in[3];  // [32,32]
  const float* b1 = (const float*)d_in[4];  // [32]
  const float* W2 = (const float*)d_in[5];  // [32,32]
  const float* b2 = (const float*)d_in[6];  // [32]
  float* out = (float*)d_out;

  const int N = in_sizes[0] / DF;
  const int E = in_sizes[1];
  const int NF = N * DF;       // elements in a feature matrix
  const int tot = E * DF;      // 51.2M (edge,channel) pairs

  // Workspace layout (floats): norm_out | norm_in | xs (reused as h1s) | agg
  float* ws = (float*)d_ws;
  float* norm_out = ws;                          // N
  float* norm_in = ws + N;                       // N
  float* xs = ws + 2 * (size_t)N;                // 32N (prescaled feats / h1s)
  float* agg = ws + 2 * (size_t)N + (size_t)NF;  // 32N (scatter accumulator)

  const int T = 256;
  const int gB = ((N + 15) / 16 + 7) / 8;  // waves of 16 rows, 8 waves/block

  // Degrees -> norms (zero first: harness poisons d_ws).
  k_zero<<<(2 * N + T - 1) / T, T, 0, stream>>>(norm_out, 2 * N);
  k_degrees<<<(E + T - 1) / T, T, 0, stream>>>(src, dst, norm_out, norm_in, E);
  k_invsqrt<<<(N + T - 1) / T, T, 0, stream>>>(norm_out, norm_in, N);

  // ---- Layer 1 ----
  k_prescale<<<(NF + T - 1) / T, T, 0, stream>>>(x, norm_out, xs, NF);
  k_zero<<<(NF + T - 1) / T, T, 0, stream>>>(agg, NF);
  k_scatter<<<(tot + T - 1) / T, T, 0, stream>>>(xs, src, dst, agg, tot);
  // h1s = relu((agg*ni) @ W1 + b1) * norm_out   (written over xs)
  k_gemm_wmma<<<gB, 256, 0, stream>>>(agg, norm_in, W1, b1, norm_out, xs, N, 1);

  // ---- Layer 2 ----
  k_zero<<<(NF + T - 1) / T, T, 0, stream>>>(agg, NF);
  k_scatter<<<(tot + T - 1) / T, T, 0, stream>>>(xs, src, dst, agg, tot);
  // y = (agg*ni) @ W2 + b2
  k_gemm_wmma<<<gB, 256, 0, stream>>>(agg, norm_in, W2, b2, nullptr, out, N, 0);
}